// ConeFilter_58609123721754
// MI455X (gfx1250) — compile-verified
//
#include <hip/hip_runtime.h>

// ConeFilter: out[b,f,t] = noisy[b,f,t] * tot_impedance[f]
// tot_impedance[f] = prod_{i=0..99} frac(f, x_i) * RHO*C/(PI*r_i^2)
//
// Memory-bound (0.125 FLOP/byte): ~263 MB HBM traffic -> ~11.3 us floor at
// 23.3 TB/s. Strategy: tiny kernel for the 257-entry product, then a
// streaming multiply using the CDNA5 async global->LDS path
// (global_load_async_to_lds_b128 + s_wait_asynccnt) and non-temporal
// 128-bit stores.

typedef float float4v __attribute__((ext_vector_type(4)));

#define F_BINS   257
#define T_LEN    4000
#define VEC_ROW  (T_LEN / 4)          // 1000 float4 per row
#define STEPS    100
#define PI_F     3.1415927f
#define C_F      343.0f
#define RHO_F    1.293f
#define SR_F     44100.0f

// ---------------------------------------------------------------------------
// Kernel A: tot_impedance[0..256] -> d_ws.  One block, trivial cost.
// ---------------------------------------------------------------------------
__global__ __launch_bounds__(288) void cone_tot_kernel(
    const float* __restrict__ x0p, const float* __restrict__ anglep,
    const float* __restrict__ depthp, float* __restrict__ tot) {
  const int f = threadIdx.x;
  if (f >= F_BINS) return;

  const float x0  = fmaxf(x0p[0], 0.0f);                      // relu(x0)
  const float dep = fmaxf(depthp[0], 0.0f);                   // relu(depth)
  const float sig = 1.0f / (1.0f + expf(-anglep[0]));         // sigmoid
  const float ang = (sig + 1.0f) * 0.5f * (PI_F * 0.5f);
  const float t   = tanf(ang);
  // freq_map = linspace(0, sr/2, 257): f * (sr/2)/256
  const float fm  = (float)f * (SR_F * 0.5f / (float)(F_BINS - 1));
  const float ds  = dep / (float)STEPS;

  float acc = 1.0f;
  for (int i = 0; i < STEPS; ++i) {
    float x     = x0 + (float)i * ds;
    float r     = t * x;
    float numer = fm * x;
    float frac  = numer / (numer + C_F);
    float imp   = frac * (RHO_F * C_F) / (PI_F * r * r);
    acc *= imp;                                  // sequential, matches prod
  }
  tot[f] = acc;
}

// ---------------------------------------------------------------------------
// Kernel B: one block per (b,f) row of 4000 floats.
// Async-stage the row into LDS (ASYNCcnt path), then multiply + NT store.
// ---------------------------------------------------------------------------
__global__ __launch_bounds__(256) void cone_scale_async_kernel(
    const float* __restrict__ noisy, const float* __restrict__ tot,
    float* __restrict__ out) {
  __shared__ float4v stage[4 * 256];             // 16 KB staging buffer

  const int   row = blockIdx.x;                  // row = b*257 + f
  const int   f   = row % F_BINS;
  const float s   = tot[f];                      // block-uniform scale

  const float4v* __restrict__ src =
      (const float4v*)(noisy + (size_t)row * T_LEN);
  float4v* __restrict__ dst = (float4v*)(out + (size_t)row * T_LEN);

  const int tid = threadIdx.x;
  // Generic LDS address: low 32 bits are the workgroup-relative LDS offset.
  unsigned lds0 = (unsigned)(unsigned long long)&stage[tid];

  // Fire all async global->LDS loads for this thread (<= 4 in flight).
#pragma unroll
  for (int k = 0; k < 4; ++k) {
    int v = tid + k * 256;
    if (v < VEC_ROW) {
      unsigned lds_addr = lds0 + (unsigned)(k * 256 * sizeof(float4v));
      const float4v* p = src + v;
      asm volatile("global_load_async_to_lds_b128 %0, %1, off"
                   :: "v"(lds_addr), "v"(p)
                   : "memory");
    }
  }
  // Wait for this wave's async transfers to land in LDS.
  asm volatile("s_wait_asynccnt 0x0" ::: "memory");

  // Each lane reads back only its own staged slots: no barrier needed.
#pragma unroll
  for (int k = 0; k < 4; ++k) {
    int v = tid + k * 256;
    if (v < VEC_ROW) {
      float4v a = stage[k * 256 + tid];          // ds_load_b128
      float4v r = a * s;
      __builtin_nontemporal_store(r, dst + v);   // global_store_b128 NT
    }
  }
}

// ---------------------------------------------------------------------------
extern "C" void kernel_launch(void* const* d_in, const int* in_sizes, int n_in,
                              void* d_out, int out_size, void* d_ws,
                              size_t ws_size, hipStream_t stream) {
  const float* noisy = (const float*)d_in[0];  // [32, 257, 4000] f32
  const float* x0    = (const float*)d_in[1];  // [1] f32
  const float* angle = (const float*)d_in[2];  // [1] f32
  const float* depth = (const float*)d_in[3];  // [1] f32
  float*       out   = (float*)d_out;          // [32, 257, 4000] f32
  float*       tot   = (float*)d_ws;           // 257 floats scratch

  const int rows = in_sizes[0] / T_LEN;        // 32 * 257 = 8224

  hipLaunchKernelGGL(cone_tot_kernel, dim3(1), dim3(288), 0, stream,
                     x0, angle, depth, tot);
  hipLaunchKernelGGL(cone_scale_async_kernel, dim3(rows), dim3(256), 0, stream,
                     noisy, tot, out);
}